// BeliefPropagation_52055003628259
// MI455X (gfx1250) — compile-verified
//
#include <hip/hip_runtime.h>
#include <hip/hip_bf16.h>
#include <math.h>

#define Q 8
#define LDS_STRIDE 130  // 128 rows + 2 pad words -> 16 cols hit distinct banks

typedef float v2f __attribute__((ext_vector_type(2)));
typedef float v8f __attribute__((ext_vector_type(8)));

// ---------------------------------------------------------------------------
// Block-wide column sum of 128 rows x 8 cols using V_WMMA_F32_16X16X4_F32
// with an all-ones A matrix: D(16x16) = ones(16x4) * B(4x16) + C, so every
// row of D is the colsum of the 4 B rows. Staging LDS is transposed
// (lds[col*130 + row]) with columns 8..15 zero-filled:
//   - B operand pair (rows r, r+1 at one column) is one ds_load_b64
//   - no per-element predication (no exec save/restore)
//   - stride 130 -> conflict-free banks across the 16 columns
// Wave 0 chains 32 accumulating WMMAs (f32-exact); lanes 0..7 emit one
// global atomic per column.
// ---------------------------------------------------------------------------
__device__ __forceinline__ void block_colsum_wmma(const float* p, float* hacc,
                                                  float* lds) {
  const int t = threadIdx.x;  // 0..127
#pragma unroll
  for (int q = 0; q < Q; ++q) lds[q * LDS_STRIDE + t] = p[q];
#pragma unroll
  for (int q = Q; q < 16; ++q) lds[q * LDS_STRIDE + t] = 0.0f;
  __syncthreads();
  if (t < 32) {  // wave 0 only; EXEC all-ones within the wave
    const int col = t & 15;
    const int half = t >> 4;
    v2f a;
    a[0] = 1.0f;
    a[1] = 1.0f;
    v8f acc = {};
    // B layout per lane: b[0] -> K = 2*half, b[1] -> K = 2*half+1 (rows r,r+1)
    const float* basep = lds + col * LDS_STRIDE + 2 * half;
#pragma unroll
    for (int k = 0; k < 32; ++k) {  // 128 rows, 4 per WMMA
      const v2f b = *reinterpret_cast<const v2f*>(basep + 4 * k);
      acc = __builtin_amdgcn_wmma_f32_16x16x4_f32(false, a, false, b, (short)0,
                                                  acc, false, false);
    }
    // D VGPR0, lanes 0..15 = row M=0, N=lane -> colsum for column `lane`
    if (t < Q) atomicAdd(&hacc[t], acc[0]);
  }
}

// ---------------------------------------------------------------------------
// params[0] = expm1(beta) = c ; params[1] = -beta * mean_w
// ---------------------------------------------------------------------------
__global__ void params_kernel(const float* __restrict__ beta,
                              float* __restrict__ params, float mean_w) {
  if (threadIdx.x == 0) {
    const float b = beta[0];
    params[0] = expm1f(b);
    params[1] = -b * mean_w;
  }
}

// h[q] = params[1] * hacc[q]; hacc[q] = 0  (ready for next accumulation)
__global__ void h_kernel(float* __restrict__ h, float* __restrict__ hacc,
                         const float* __restrict__ params) {
  const int q = threadIdx.x;
  if (q < Q) {
    const float v = hacc[q];
    h[q] = params[1] * v;
    hacc[q] = 0.0f;
  }
}

// ---------------------------------------------------------------------------
// psi = psi0 / rowsum(psi0); colsum(psi) -> hacc (for initial h)
// ---------------------------------------------------------------------------
__global__ void init_psi_kernel(const float* __restrict__ psi0,
                                float* __restrict__ psi,
                                float* __restrict__ hacc, int N) {
  __shared__ float lds[16 * LDS_STRIDE];
  const int tid = threadIdx.x;
  const int n = blockIdx.x * 128 + tid;
  float p[Q];
  if (n < N) {
    const float4* p4 = reinterpret_cast<const float4*>(psi0 + (size_t)n * Q);
    const float4 x = p4[0], y = p4[1];
    p[0] = x.x; p[1] = x.y; p[2] = x.z; p[3] = x.w;
    p[4] = y.x; p[5] = y.y; p[6] = y.z; p[7] = y.w;
    float s = 0.0f;
#pragma unroll
    for (int q = 0; q < Q; ++q) s += p[q];
    const float inv = 1.0f / s;
#pragma unroll
    for (int q = 0; q < Q; ++q) p[q] *= inv;
    float4* o4 = reinterpret_cast<float4*>(psi + (size_t)n * Q);
    o4[0] = make_float4(p[0], p[1], p[2], p[3]);
    o4[1] = make_float4(p[4], p[5], p[6], p[7]);
  } else {
#pragma unroll
    for (int q = 0; q < Q; ++q) p[q] = 0.0f;
  }
  block_colsum_wmma(p, hacc, lds);
}

// ---------------------------------------------------------------------------
// msg = msg0 / rowsum(msg0); scatter log1p(c*msg) into S[dst] (seg-sum)
// ---------------------------------------------------------------------------
__global__ void init_msg_kernel(const float* __restrict__ msg0,
                                float* __restrict__ msg, float* __restrict__ S,
                                const float* __restrict__ params,
                                const int* __restrict__ dst, int M) {
  __shared__ float sc;
  const int tid = threadIdx.x;
  if (tid == 0) sc = params[0];
  __syncthreads();
  const int e = blockIdx.x * blockDim.x + tid;
  if (e >= M) return;
  const float c = sc;
  float p[Q];
  const float4* p4 = reinterpret_cast<const float4*>(msg0 + (size_t)e * Q);
  const float4 x = p4[0], y = p4[1];
  p[0] = x.x; p[1] = x.y; p[2] = x.z; p[3] = x.w;
  p[4] = y.x; p[5] = y.y; p[6] = y.z; p[7] = y.w;
  float s = 0.0f;
#pragma unroll
  for (int q = 0; q < Q; ++q) s += p[q];
  const float inv = 1.0f / s;
#pragma unroll
  for (int q = 0; q < Q; ++q) p[q] *= inv;
  float4* o4 = reinterpret_cast<float4*>(msg + (size_t)e * Q);
  o4[0] = make_float4(p[0], p[1], p[2], p[3]);
  o4[1] = make_float4(p[4], p[5], p[6], p[7]);
  const int d = dst[e];
  float* Sd = S + (size_t)d * Q;
#pragma unroll
  for (int q = 0; q < Q; ++q) atomicAdd(&Sd[q], log1pf(c * p[q]));
}

// ---------------------------------------------------------------------------
// Message update (one thread per directed edge):
//   logits = h + S_cur[src] - log1p(c*msg_cur[rev])
//   new_msg = softmax(logits); scatter log1p(c*new_msg) into S_next[dst]
//   (last iter) diff = max|new_msg - msg_cur|
// ---------------------------------------------------------------------------
__global__ void bp_msg_kernel(const float* __restrict__ msg_cur,
                              float* __restrict__ msg_next,
                              const float* __restrict__ S_cur,
                              float* __restrict__ S_next,
                              const float* __restrict__ h,
                              const float* __restrict__ params,
                              const int* __restrict__ src,
                              const int* __restrict__ dst,
                              const int* __restrict__ rev, int M,
                              int computeDiff,
                              unsigned int* __restrict__ diffOut) {
  __shared__ float sh[Q];
  __shared__ float sc;
  __shared__ float sred[256];
  const int tid = threadIdx.x;
  if (tid < Q) sh[tid] = h[tid];
  if (tid == 0) sc = params[0];
  __syncthreads();
  const int e = blockIdx.x * blockDim.x + tid;
  float dm = 0.0f;
  if (e < M) {
    const float c = sc;
    const int s = src[e];
    const int d = dst[e];
    const int r = rev[e];
    // gather S_cur[src] (32B, L2-resident)
    float Ss[Q];
    {
      const float4* p4 = reinterpret_cast<const float4*>(S_cur + (size_t)s * Q);
      const float4 x = p4[0], y = p4[1];
      Ss[0] = x.x; Ss[1] = x.y; Ss[2] = x.z; Ss[3] = x.w;
      Ss[4] = y.x; Ss[5] = y.y; Ss[6] = y.z; Ss[7] = y.w;
    }
    // gather msg_cur[rev]; recompute its log-factor (cheaper than storing log_f)
    float mr[Q];
    {
      const float4* p4 =
          reinterpret_cast<const float4*>(msg_cur + (size_t)r * Q);
      const float4 x = p4[0], y = p4[1];
      mr[0] = x.x; mr[1] = x.y; mr[2] = x.z; mr[3] = x.w;
      mr[4] = y.x; mr[5] = y.y; mr[6] = y.z; mr[7] = y.w;
    }
    float lg[Q];
#pragma unroll
    for (int q = 0; q < Q; ++q) lg[q] = sh[q] + Ss[q] - log1pf(c * mr[q]);
    float mx = lg[0];
#pragma unroll
    for (int q = 1; q < Q; ++q) mx = fmaxf(mx, lg[q]);
    float ex[Q];
    float sum = 0.0f;
#pragma unroll
    for (int q = 0; q < Q; ++q) {
      ex[q] = expf(lg[q] - mx);
      sum += ex[q];
    }
    const float inv = 1.0f / sum;
    float nm[Q];
#pragma unroll
    for (int q = 0; q < Q; ++q) nm[q] = ex[q] * inv;
    float4* o4 = reinterpret_cast<float4*>(msg_next + (size_t)e * Q);
    o4[0] = make_float4(nm[0], nm[1], nm[2], nm[3]);
    o4[1] = make_float4(nm[4], nm[5], nm[6], nm[7]);
    // segment-sum for next iteration (S2(t) == S(t+1))
    float* Sd = S_next + (size_t)d * Q;
#pragma unroll
    for (int q = 0; q < Q; ++q) atomicAdd(&Sd[q], log1pf(c * nm[q]));
    if (computeDiff) {
      const float4* p4 =
          reinterpret_cast<const float4*>(msg_cur + (size_t)e * Q);
      const float4 x = p4[0], y = p4[1];
      float om[Q];
      om[0] = x.x; om[1] = x.y; om[2] = x.z; om[3] = x.w;
      om[4] = y.x; om[5] = y.y; om[6] = y.z; om[7] = y.w;
#pragma unroll
      for (int q = 0; q < Q; ++q) dm = fmaxf(dm, fabsf(nm[q] - om[q]));
    }
  }
  if (computeDiff) {  // uniform predicate -> barriers are safe
    sred[tid] = dm;
    __syncthreads();
    for (int st = 128; st > 0; st >>= 1) {
      if (tid < st) sred[tid] = fmaxf(sred[tid], sred[tid + st]);
      __syncthreads();
    }
    if (tid == 0) atomicMax(diffOut, __float_as_uint(sred[0]));
  }
}

// ---------------------------------------------------------------------------
// psi = softmax(h + S2); colsum(psi) -> hacc (feeds next h)
// ---------------------------------------------------------------------------
__global__ void psi_kernel(const float* __restrict__ S2,
                           const float* __restrict__ h,
                           float* __restrict__ psi, float* __restrict__ hacc,
                           int N) {
  __shared__ float sh[Q];
  __shared__ float lds[16 * LDS_STRIDE];
  const int tid = threadIdx.x;
  if (tid < Q) sh[tid] = h[tid];
  __syncthreads();
  const int n = blockIdx.x * 128 + tid;
  float p[Q];
  if (n < N) {
    float l[Q];
    {
      const float4* p4 = reinterpret_cast<const float4*>(S2 + (size_t)n * Q);
      const float4 x = p4[0], y = p4[1];
      l[0] = sh[0] + x.x; l[1] = sh[1] + x.y;
      l[2] = sh[2] + x.z; l[3] = sh[3] + x.w;
      l[4] = sh[4] + y.x; l[5] = sh[5] + y.y;
      l[6] = sh[6] + y.z; l[7] = sh[7] + y.w;
    }
    float mx = l[0];
#pragma unroll
    for (int q = 1; q < Q; ++q) mx = fmaxf(mx, l[q]);
    float sum = 0.0f;
#pragma unroll
    for (int q = 0; q < Q; ++q) {
      p[q] = expf(l[q] - mx);
      sum += p[q];
    }
    const float inv = 1.0f / sum;
#pragma unroll
    for (int q = 0; q < Q; ++q) p[q] *= inv;
    float4* o4 = reinterpret_cast<float4*>(psi + (size_t)n * Q);
    o4[0] = make_float4(p[0], p[1], p[2], p[3]);
    o4[1] = make_float4(p[4], p[5], p[6], p[7]);
  } else {
#pragma unroll
    for (int q = 0; q < Q; ++q) p[q] = 0.0f;
  }
  block_colsum_wmma(p, hacc, lds);
}

// ---------------------------------------------------------------------------
extern "C" void kernel_launch(void* const* d_in, const int* in_sizes, int n_in,
                              void* d_out, int out_size, void* d_ws,
                              size_t ws_size, hipStream_t stream) {
  (void)n_in;
  (void)out_size;
  (void)ws_size;
  const float* beta = (const float*)d_in[0];
  const float* psi0 = (const float*)d_in[1];
  const float* msg0 = (const float*)d_in[2];
  const int* src = (const int*)d_in[3];
  const int* dst = (const int*)d_in[4];
  const int* rev = (const int*)d_in[5];
  const int N = in_sizes[1] / Q;
  const int M = in_sizes[3];          // 2E directed messages
  const int NUM_ITER = 5;             // reference max_num_iter (fixed)
  const float mean_w = (float)((double)M / ((double)N * (double)N));

  // workspace carve-out (256B aligned)
  char* w = (char*)d_ws;
  size_t off = 0;
  auto take = [&](size_t bytes) -> char* {
    char* p = w + off;
    off = (off + bytes + 255) & ~(size_t)255;
    return p;
  };
  float* msgB = (float*)take((size_t)M * Q * sizeof(float));
  float* S_A = (float*)take((size_t)N * Q * sizeof(float));
  float* S_B = (float*)take((size_t)N * Q * sizeof(float));
  float* hacc = (float*)take(Q * sizeof(float));
  float* hbuf = (float*)take(Q * sizeof(float));
  float* params = (float*)take(2 * sizeof(float));

  float* msg_out = (float*)d_out;
  float* psi_out = msg_out + (size_t)M * Q;
  float* diff_out = psi_out + (size_t)N * Q;

  hipMemsetAsync(S_A, 0, (size_t)N * Q * sizeof(float), stream);
  hipMemsetAsync(hacc, 0, Q * sizeof(float), stream);
  hipMemsetAsync(diff_out, 0, sizeof(float), stream);

  params_kernel<<<1, 32, 0, stream>>>(beta, params, mean_w);
  init_psi_kernel<<<(N + 127) / 128, 128, 0, stream>>>(psi0, psi_out, hacc, N);
  h_kernel<<<1, 32, 0, stream>>>(hbuf, hacc, params);
  init_msg_kernel<<<(M + 255) / 256, 256, 0, stream>>>(msg0, msgB, S_A, params,
                                                       dst, M);

  // ping-pong: initial msg in ws; odd #iters lands the final msg in d_out
  float* mcur = msgB;
  float* mnext = msg_out;
  float* Scur = S_A;
  float* Snext = S_B;
  for (int t = 0; t < NUM_ITER; ++t) {
    hipMemsetAsync(Snext, 0, (size_t)N * Q * sizeof(float), stream);
    const int cd = (t == NUM_ITER - 1) ? 1 : 0;
    bp_msg_kernel<<<(M + 255) / 256, 256, 0, stream>>>(
        mcur, mnext, Scur, Snext, hbuf, params, src, dst, rev, M, cd,
        (unsigned int*)diff_out);
    psi_kernel<<<(N + 127) / 128, 128, 0, stream>>>(Snext, hbuf, psi_out, hacc,
                                                    N);
    h_kernel<<<1, 32, 0, stream>>>(hbuf, hacc, params);
    float* tm = mcur; mcur = mnext; mnext = tm;
    float* ts = Scur; Scur = Snext; Snext = ts;
  }
  if (mcur != msg_out) {  // only if NUM_ITER were even
    hipMemcpyAsync(msg_out, mcur, (size_t)M * Q * sizeof(float),
                   hipMemcpyDeviceToDevice, stream);
  }
}